// LinearDecayEmbedding_45037027066297
// MI455X (gfx1250) — compile-verified
//
#include <hip/hip_runtime.h>

// LinearDecayEmbedding for MI455X (gfx1250, wave32).
//
// out[b,s, k*Q + (q-1)] = clip(1 - |k - r|/4, 0)  for q = question_ids[b,s] > 0
// everything else = 0.
//
// Pure store-bandwidth problem: 256 MB fp32 output. One 256-thread (8-wave)
// block per (b,s) row; thread t (< 250) owns columns [4t, 4t+4) of the Q-block
// and emits one non-temporal global_store_b128 per category k (K=5, unrolled).
// All weights are block-uniform (SGPRs); per-lane work is 4 compares + 4
// selects per store.

namespace {

constexpr int Q     = 1000;    // questions
constexpr int K     = 5;       // categories
constexpr int ROW   = K * Q;   // 5000 floats per (b,s) row
constexpr int QCHK  = Q / 4;   // 250 float4 chunks per Q-block

typedef __attribute__((ext_vector_type(4))) float v4f;

__global__ __launch_bounds__(256)
void linear_decay_embedding_kernel(const int* __restrict__ qids,
                                   const int* __restrict__ resps,
                                   float* __restrict__ out,
                                   int n_rows) {
    const int row = blockIdx.x;
    if (row >= n_rows) return;

    // Block-uniform -> scalar loads (s_load_b32) + SALU.
    const int   qid = qids[row];
    const int   r   = resps[row];
    const int   col = qid - 1;          // target column within a Q-block; -1 if padding
    const float rf  = (float)r;

    // All K weights are uniform: live in SGPRs.
    float w[K];
#pragma unroll
    for (int k = 0; k < K; ++k) {
        w[k] = fmaxf(0.0f, 1.0f - fabsf((float)k - rf) * 0.25f);
    }

    const int t = threadIdx.x;
    if (t >= QCHK) return;              // 250 active lanes per block
    const int qcol = t * 4;             // column within the Q-block, 0..996

    // Column-match predicates: computed once, reused for all 5 k-segments.
    const bool m0 = (qcol + 0 == col);
    const bool m1 = (qcol + 1 == col);
    const bool m2 = (qcol + 2 == col);
    const bool m3 = (qcol + 3 == col);

    float* __restrict__ row_out = out + (size_t)row * (size_t)ROW + qcol;

#pragma unroll
    for (int k = 0; k < K; ++k) {
        v4f v;
        v.x = m0 ? w[k] : 0.0f;
        v.y = m1 ? w[k] : 0.0f;
        v.z = m2 ? w[k] : 0.0f;
        v.w = m3 ? w[k] : 0.0f;
        // Streaming NT store: 256 MB output > 192 MB L2, never re-read here.
        __builtin_nontemporal_store(v, reinterpret_cast<v4f*>(row_out + k * Q));
    }
}

} // namespace

extern "C" void kernel_launch(void* const* d_in, const int* in_sizes, int n_in,
                              void* d_out, int out_size, void* d_ws, size_t ws_size,
                              hipStream_t stream) {
    const int* qids  = (const int*)d_in[0];   // question_ids, (B,S) int32
    const int* resps = (const int*)d_in[1];   // responses,    (B,S) int32
    float*     out   = (float*)d_out;         // (B,S,K*Q) fp32

    const int n_rows = in_sizes[0];           // B*S = 12800

    dim3 grid((unsigned)n_rows);
    dim3 block(256);                          // 8 wave32 waves
    linear_decay_embedding_kernel<<<grid, block, 0, stream>>>(qids, resps, out, n_rows);
}